// TetrisMaster2_18193481466484
// MI455X (gfx1250) — compile-verified
//
#include <hip/hip_runtime.h>
#include <hip/hip_bf16.h>
#include <math.h>

// MI455X / gfx1250, wave32. Latency-bound sequential net -> one persistent
// workgroup (128 threads = 4 waves = 1 wave/SIMD32), activations in LDS,
// all GEMVs via V_WMMA_F32_16X16X4_F32 (fp32 keeps reference precision).
//
// Latency hiding: per row tile, all 25 A-loads of a chunk are issued into a
// fully-unrolled register array (clause + staged s_wait_loadcnt), then the
// 25 chained WMMAs run; two row tiles are processed concurrently per wave
// (independent accumulator chains, shared B registers).
//
// Padding: padded rows clamp to the last valid row (in-bounds loads, no
// exec-masked branches); padded outputs land in LDS slots never read.

typedef __attribute__((ext_vector_type(2))) float v2f;
typedef __attribute__((ext_vector_type(8))) float v8f;

#define NLAYERS 256

__device__ __forceinline__ v8f wmma4(v2f a, v2f b, v8f c) {
  return __builtin_amdgcn_wmma_f32_16x16x4_f32(false, a, false, b, (short)0, c,
                                               false, false);
}

// y[0..rows) = act(W[rows x K] * vin + bias).  K % 100 == 0 (chunk = 25 k-tiles).
// A tile (16x4 f32): lane L holds row rbase+(L&15), K-offsets {0,1} (L<16) or {2,3}.
// B tile (4x16 f32): v broadcast across all 16 N columns -> every D column = y.
// C/D (16x16 f32): VGPR g = row rbase+g (lanes 0-15) / rbase+g+8 (lanes 16-31).
template <int K, bool DO_RELU>
__device__ __forceinline__ void wave_gemv(
    const float* __restrict__ W, const float* __restrict__ bias,
    const float* vin, float* vout, int rows, int ld,
    int wave, int nwaves, int lane, const float* __restrict__ Wnext)
{
  constexpr int NT = K / 4;   // k-tiles
  constexpr int CH = 25;      // k-tiles per chunk
  static_assert(NT % CH == 0, "K must be a multiple of 100");

  const int rowTiles = (rows + 15) >> 4;
  const int m     = lane & 15;
  const int khalf = lane >> 4;       // 0 or 1
  const int koff  = khalf * 2;

  int rt = wave;
  // ---- paired row tiles: two independent accumulator chains, shared B ----
  for (; rt + nwaves < rowTiles; rt += 2 * nwaves) {
    const int rb0 = rt << 4;
    const int rb1 = (rt + nwaves) << 4;
    int r0 = rb0 + m; if (r0 > rows - 1) r0 = rows - 1;
    int r1 = rb1 + m; if (r1 > rows - 1) r1 = rows - 1;
    const float* wp0 = W + (size_t)r0 * (size_t)ld + koff;
    const float* wp1 = W + (size_t)r1 * (size_t)ld + koff;

    if (Wnext) {
      __builtin_prefetch(Wnext + (size_t)r0 * (size_t)ld + khalf * (K >> 1), 0, 1);
      __builtin_prefetch(Wnext + (size_t)r1 * (size_t)ld + khalf * (K >> 1), 0, 1);
    }

    v8f acc0, acc1;
#pragma unroll
    for (int g = 0; g < 8; ++g) {
      int q0 = rb0 + g + khalf * 8; acc0[g] = bias[q0 < rows ? q0 : rows - 1];
      int q1 = rb1 + g + khalf * 8; acc1[g] = bias[q1 < rows ? q1 : rows - 1];
    }

    for (int c = 0; c < NT; c += CH) {
      v2f a0[CH], a1[CH], bb[CH];
#pragma unroll
      for (int t = 0; t < CH; ++t) {           // batch-issue all loads (clause)
        a0[t] = *(const v2f*)(wp0 + 4 * (c + t));
        a1[t] = *(const v2f*)(wp1 + 4 * (c + t));
        bb[t] = *(const v2f*)(vin + 4 * (c + t) + koff);
      }
#pragma unroll
      for (int t = 0; t < CH; ++t) {           // two interleaved WMMA chains
        acc0 = wmma4(a0[t], bb[t], acc0);
        acc1 = wmma4(a1[t], bb[t], acc1);
      }
    }

    if (DO_RELU) {
#pragma unroll
      for (int g = 0; g < 8; ++g) {
        acc0[g] = fmaxf(acc0[g], 0.0f);
        acc1[g] = fmaxf(acc1[g], 0.0f);
      }
    }
    if (m == 0) {  // lane 0 -> rows rb+0..7, lane 16 -> rows rb+8..15
#pragma unroll
      for (int g = 0; g < 8; ++g) {
        vout[rb0 + khalf * 8 + g] = acc0[g];
        vout[rb1 + khalf * 8 + g] = acc1[g];
      }
    }
  }
  // ---- leftover single row tiles ----
  for (; rt < rowTiles; rt += nwaves) {
    const int rb = rt << 4;
    int r = rb + m; if (r > rows - 1) r = rows - 1;
    const float* wp = W + (size_t)r * (size_t)ld + koff;

    if (Wnext)
      __builtin_prefetch(Wnext + (size_t)r * (size_t)ld + khalf * (K >> 1), 0, 1);

    v8f acc;
#pragma unroll
    for (int g = 0; g < 8; ++g) {
      int q = rb + g + khalf * 8; acc[g] = bias[q < rows ? q : rows - 1];
    }

    for (int c = 0; c < NT; c += CH) {
      v2f a[CH], bb[CH];
#pragma unroll
      for (int t = 0; t < CH; ++t) {
        a[t]  = *(const v2f*)(wp + 4 * (c + t));
        bb[t] = *(const v2f*)(vin + 4 * (c + t) + koff);
      }
#pragma unroll
      for (int t = 0; t < CH; ++t) acc = wmma4(a[t], bb[t], acc);
    }

    if (DO_RELU) {
#pragma unroll
      for (int g = 0; g < 8; ++g) acc[g] = fmaxf(acc[g], 0.0f);
    }
    if (m == 0) {
#pragma unroll
      for (int g = 0; g < 8; ++g) vout[rb + khalf * 8 + g] = acc[g];
    }
  }
}

__global__ __launch_bounds__(128)
void tetris_net_kernel(const float* __restrict__ b,        const float* __restrict__ n,
                       const float* __restrict__ conv_w,   const float* __restrict__ conv_b,
                       const float* __restrict__ deconv_w, const float* __restrict__ deconv_b,
                       const float* __restrict__ mid_w1,   const float* __restrict__ mid_b1,
                       const float* __restrict__ mid_w2,   const float* __restrict__ mid_b2,
                       const float* __restrict__ mid_w3,   const float* __restrict__ mid_b3,
                       const float* __restrict__ out_w0,   const float* __restrict__ out_b0,
                       const float* __restrict__ out_ws,   const float* __restrict__ out_bs,
                       const float* __restrict__ out_wf,   const float* __restrict__ out_bf,
                       float* __restrict__ out)
{
  __shared__ float x0[200];      // input board [20][10]
  __shared__ float y4[200];      // relu(pool(conv)) [4][10][5]
  __shared__ float h1[100];
  __shared__ float h2[1008];     // 1000 padded to 63 row tiles
  __shared__ float h3[16];
  __shared__ float u[200];       // deconv+skip [20][10] flattened
  __shared__ float vbuf[2][128]; // 100 padded to 7 row tiles (112)
  __shared__ float lg[16];       // logits padded

  const int tid  = threadIdx.x;
  const int lane = tid & 31;
  const int wave = tid >> 5;

  if (tid < 200) x0[tid] = b[tid];
  if (tid < 100) {
    float t = n[0] * (1.0f / 6.0f);
    h1[tid] = fmaxf(mid_w1[tid] * t + mid_b1[tid], 0.0f);
  }
  __syncthreads();

  // --- Conv2d(1,4,k=2,p=1) -> MaxPool(2) -> ReLU : 200 outputs, 1 thread each ---
  if (tid < 200) {
    int c = tid / 50, rem = tid % 50, pi = rem / 5, pj = rem % 5;
    float cb = conv_b[c];
    float mx = -INFINITY;
#pragma unroll
    for (int di = 0; di < 2; ++di)
#pragma unroll
      for (int dj = 0; dj < 2; ++dj) {
        int oi = 2 * pi + di, oj = 2 * pj + dj;  // conv grid 21x11; rows 0..19/cols 0..9 used
        float s = cb;
#pragma unroll
        for (int kh = 0; kh < 2; ++kh)
#pragma unroll
          for (int kw = 0; kw < 2; ++kw) {
            int xi = oi - 1 + kh, xj = oj - 1 + kw;
            float xv = (xi >= 0 && xi < 20 && xj >= 0 && xj < 10) ? x0[xi * 10 + xj] : 0.0f;
            s += conv_w[c * 4 + kh * 2 + kw] * xv;
          }
        mx = fmaxf(mx, s);
      }
    y4[tid] = fmaxf(mx, 0.0f);
  }
  __syncthreads();

  // --- mid MLP: h2 = relu(mid_w2 @ h1 + b2)  [1000 x 100] (all 4 waves) ---
  wave_gemv<100, true>(mid_w2, mid_b2, h1, h2, 1000, 100, wave, 4, lane, nullptr);
  __syncthreads();

  // --- h3 = relu(mid_w3 @ h2 + b3)  [4 x 1000] (wave 0) ---
  if (wave == 0)
    wave_gemv<1000, true>(mid_w3, mid_b3, h2, h3, 4, 1000, 0, 1, lane, nullptr);
  __syncthreads();

  // --- ConvTranspose2d(4,1,k=2,s=2) -> ReLU -> +skip.  Each out pixel has
  //     exactly one contribution: u(i,j) = relu(db + sum_c (y+h3)[c][i/2][j/2]*w[c][i&1][j&1]) + x0
  if (tid < 200) {
    int i = tid / 10, j = tid % 10;
    float s = deconv_b[0];
#pragma unroll
    for (int c = 0; c < 4; ++c)
      s += (y4[c * 50 + (i >> 1) * 5 + (j >> 1)] + h3[c]) *
           deconv_w[c * 4 + (i & 1) * 2 + (j & 1)];
    u[tid] = fmaxf(s, 0.0f) + x0[tid];
  }
  __syncthreads();

  // --- out head: v = relu(out_w0 @ u + b0)  [100 x 200] ---
  wave_gemv<200, true>(out_w0, out_b0, u, vbuf[0], 100, 200, wave, 4, lane, out_ws);
  __syncthreads();

  // --- 256 x (Linear(100,100) + ReLU), ping-pong LDS buffers ---
  int cur = 0;
  for (int l = 0; l < NLAYERS; ++l) {
    const float* W  = out_ws + (size_t)l * 10000;
    const float* Wn = (l + 1 < NLAYERS) ? (W + 10000) : nullptr;
    wave_gemv<100, true>(W, out_bs + l * 100, vbuf[cur], vbuf[cur ^ 1],
                         100, 100, wave, 4, lane, Wn);
    cur ^= 1;
    __syncthreads();
  }

  // --- logits = out_wf @ v + out_bf  [4 x 100] (wave 0), then softmax ---
  if (wave == 0)
    wave_gemv<100, false>(out_wf, out_bf, vbuf[cur], lg, 4, 100, 0, 1, lane, nullptr);
  __syncthreads();

  if (tid == 0) {
    float m = fmaxf(fmaxf(lg[0], lg[1]), fmaxf(lg[2], lg[3]));
    float e0 = __expf(lg[0] - m), e1 = __expf(lg[1] - m);
    float e2 = __expf(lg[2] - m), e3 = __expf(lg[3] - m);
    float s = e0 + e1 + e2 + e3;
    out[0] = e0 / s; out[1] = e1 / s; out[2] = e2 / s; out[3] = e3 / s;
  }
}

extern "C" void kernel_launch(void* const* d_in, const int* in_sizes, int n_in,
                              void* d_out, int out_size, void* d_ws, size_t ws_size,
                              hipStream_t stream) {
  (void)in_sizes; (void)n_in; (void)out_size; (void)d_ws; (void)ws_size;
  const float* b        = (const float*)d_in[0];
  const float* n        = (const float*)d_in[1];
  const float* conv_w   = (const float*)d_in[2];
  const float* conv_b   = (const float*)d_in[3];
  const float* deconv_w = (const float*)d_in[4];
  const float* deconv_b = (const float*)d_in[5];
  const float* mid_w1   = (const float*)d_in[6];
  const float* mid_b1   = (const float*)d_in[7];
  const float* mid_w2   = (const float*)d_in[8];
  const float* mid_b2   = (const float*)d_in[9];
  const float* mid_w3   = (const float*)d_in[10];
  const float* mid_b3   = (const float*)d_in[11];
  const float* out_w0   = (const float*)d_in[12];
  const float* out_b0   = (const float*)d_in[13];
  const float* out_ws   = (const float*)d_in[14];
  const float* out_bs   = (const float*)d_in[15];
  const float* out_wf   = (const float*)d_in[16];
  const float* out_bf   = (const float*)d_in[17];

  tetris_net_kernel<<<1, 128, 0, stream>>>(
      b, n, conv_w, conv_b, deconv_w, deconv_b,
      mid_w1, mid_b1, mid_w2, mid_b2, mid_w3, mid_b3,
      out_w0, out_b0, out_ws, out_bs, out_wf, out_bf,
      (float*)d_out);
}